// RMILoss_63153199120587
// MI455X (gfx1250) — compile-verified
//
#include <hip/hip_runtime.h>
#include <hip/hip_bf16.h>
#include <math.h>

typedef __attribute__((ext_vector_type(2))) float v2f;
typedef __attribute__((ext_vector_type(8))) float v8f;

#define NCLS   19
#define NB     4
#define HIN    512
#define WIN    512
#define PH     171                 // pooled H/W: floor((512+2-3)/3)+1
#define NHW    169                 // PH - RMI_RADIUS + 1
#define MTOT   (NHW * NHW)         // 28561 valid samples per (n,c)
#define ROWPAD 172                 // padded row length (multiple of 4)
#define MPAD   (170 * ROWPAD)      // 29240: 170 padded rows (even #chunks)
#define NCPROB (NB * NCLS)         // 76 independent covariance problems

// ---------------------------------------------------------------------------
// K1: fused softmax + valid-mask + one-hot + 3x3/s3 avg-pool (count_include_pad)
// One thread per pooled output cell; logits read exactly once (windows are
// non-overlapping at stride 3 / pad 1).
// ---------------------------------------------------------------------------
__global__ __launch_bounds__(256)
void rmi_softmax_pool(const float* __restrict__ logits,
                      const int*   __restrict__ label,
                      float*       __restrict__ pooledP,
                      float*       __restrict__ pooledO)
{
    int idx = blockIdx.x * 256 + threadIdx.x;
    const int total = NB * PH * PH;
    if (idx >= total) return;
    int pw = idx % PH;
    int t  = idx / PH;
    int ph = t % PH;
    int n  = t / PH;

    float sumP[NCLS], sumO[NCLS];
#pragma unroll
    for (int c = 0; c < NCLS; ++c) { sumP[c] = 0.f; sumO[c] = 0.f; }

    const size_t HW  = (size_t)HIN * WIN;
    const float* lgb = logits + (size_t)n * NCLS * HW;
    const int*   lab = label  + (size_t)n * HW;

    for (int dy = 0; dy < 3; ++dy) {
        int row = 3 * ph - 1 + dy;
        if (row < 0 || row >= HIN) continue;
        for (int dx = 0; dx < 3; ++dx) {
            int col = 3 * pw - 1 + dx;
            if (col < 0 || col >= WIN) continue;
            int off = row * WIN + col;
            int l = lab[off];
            if (l < 0 || l >= NCLS || l == 255) continue;  // invalid -> both masked to 0
            float v[NCLS];
            float mx = -1e30f;
#pragma unroll
            for (int c = 0; c < NCLS; ++c) {
                v[c] = lgb[(size_t)c * HW + off];
                mx = fmaxf(mx, v[c]);
            }
            float se = 0.f;
#pragma unroll
            for (int c = 0; c < NCLS; ++c) { v[c] = __expf(v[c] - mx); se += v[c]; }
            float inv = 1.0f / se;
#pragma unroll
            for (int c = 0; c < NCLS; ++c) {
                sumP[c] += v[c] * inv;
                sumO[c] += (c == l) ? 1.f : 0.f;
            }
        }
    }
    const float inv9 = 1.f / 9.f;
    size_t ob = ((size_t)n * NCLS) * (PH * PH) + (size_t)ph * PH + pw;
#pragma unroll
    for (int c = 0; c < NCLS; ++c) {
        pooledP[ob + (size_t)c * (PH * PH)] = sumP[c] * inv9;
        pooledO[ob + (size_t)c * (PH * PH)] = sumO[c] * inv9;
    }
}

// ---------------------------------------------------------------------------
// K2: per-(n,c) second-moment matrices via V_WMMA_F32_16X16X4_F32.
// Rows 0..8 = the 9 shifted 169x169 views, row 9 = ones (row sums for free),
// rows 10..15 = zero.  A and B operands share registers (X·Xᵀ layout trick).
// Sample enumeration is padded to 172 cols x 170 rows so a K=4 chunk never
// crosses a row: (i,j) advance incrementally (no integer division) and each
// lane's 2 elements are contiguous (mergeable into a b64 load).
// Two K-chunks per iteration into two accumulator sets (6 WMMAs/iter) to
// double the RAW distance on the accumulate chain.
// ---------------------------------------------------------------------------
__device__ __forceinline__ void load_pair(const float* __restrict__ bP,
                                          const float* __restrict__ bL,
                                          int i, int jj, int dy, int dx,
                                          float& p0, float& p1,
                                          float& l0, float& l1)
{
    // i: sample row (valid < NHW); jj: sample col of first element (jj+1 second)
    if (i < NHW) {
        int row = (i + dy) * PH + dx;
        if (jj + 1 < NHW) {           // both valid & contiguous
            p0 = bP[row + jj]; p1 = bP[row + jj + 1];
            l0 = bL[row + jj]; l1 = bL[row + jj + 1];
        } else if (jj < NHW) {        // last column
            p0 = bP[row + jj];
            l0 = bL[row + jj];
        }
    }
}

__global__ __launch_bounds__(128)
void rmi_cov_wmma(const float* __restrict__ pooledP,
                  const float* __restrict__ pooledO,
                  float*       __restrict__ results)
{
    const int nc   = blockIdx.x;          // 0..75
    const int lane = threadIdx.x & 31;
    const int wave = threadIdx.x >> 5;    // 4 waves split the K dimension
    const int r    = lane & 15;           // matrix row this lane feeds
    const int koff = (lane >> 4) << 1;    // lanes 0-15: K+0,1 ; lanes 16-31: K+2,3

    const float* baseP = pooledP + (size_t)nc * (PH * PH);
    const float* baseL = pooledO + (size_t)nc * (PH * PH);
    const int dy = r / 3, dx = r % 3;
    const bool isData = (r < 9);
    const bool isOnes = (r == 9);

    v8f aLL0 = {0.f,0.f,0.f,0.f,0.f,0.f,0.f,0.f};
    v8f aPP0 = {0.f,0.f,0.f,0.f,0.f,0.f,0.f,0.f};
    v8f aPL0 = {0.f,0.f,0.f,0.f,0.f,0.f,0.f,0.f};
    v8f aLL1 = {0.f,0.f,0.f,0.f,0.f,0.f,0.f,0.f};
    v8f aPP1 = {0.f,0.f,0.f,0.f,0.f,0.f,0.f,0.f};
    v8f aPL1 = {0.f,0.f,0.f,0.f,0.f,0.f,0.f,0.f};

    // m (chunk-A base) walks wave*8, wave*8+32, ... ; (i,j) track m incrementally
    int i = 0;
    int j = wave * 8;                      // < ROWPAD, so row 0
    for (int m = wave * 8; m < MPAD; m += 32) {
        // chunk B base = m + 4; may roll into the next padded row (43 odd chunks/row)
        int iB = i, jB = j + 4;
        if (jB >= ROWPAD) { jB -= ROWPAD; ++iB; }

        float l0 = 0.f, l1 = 0.f, p0 = 0.f, p1 = 0.f;
        float l2 = 0.f, l3 = 0.f, p2 = 0.f, p3 = 0.f;
        if (isData) {
            load_pair(baseP, baseL, i,  j  + koff, dy, dx, p0, p1, l0, l1);
            load_pair(baseP, baseL, iB, jB + koff, dy, dx, p2, p3, l2, l3);
        } else if (isOnes) {
            l0 = l1 = p0 = p1 = 1.f;       // padded positions harmless: data rows 0
            l2 = l3 = p2 = p3 = 1.f;
        }
        v2f A0 = {l0, l1}, P0 = {p0, p1};
        v2f A1 = {l2, l3}, P1 = {p2, p3};
        aLL0 = __builtin_amdgcn_wmma_f32_16x16x4_f32(false, A0, false, A0, (short)0, aLL0, false, false);
        aPP0 = __builtin_amdgcn_wmma_f32_16x16x4_f32(false, P0, false, P0, (short)0, aPP0, false, false);
        aPL0 = __builtin_amdgcn_wmma_f32_16x16x4_f32(false, P0, false, A0, (short)0, aPL0, false, false);
        aLL1 = __builtin_amdgcn_wmma_f32_16x16x4_f32(false, A1, false, A1, (short)0, aLL1, false, false);
        aPP1 = __builtin_amdgcn_wmma_f32_16x16x4_f32(false, P1, false, P1, (short)0, aPP1, false, false);
        aPL1 = __builtin_amdgcn_wmma_f32_16x16x4_f32(false, P1, false, A1, (short)0, aPL1, false, false);

        j += 32;
        if (j >= ROWPAD) { j -= ROWPAD; ++i; }
    }
    aLL0 += aLL1;                          // elementwise: layouts identical
    aPP0 += aPP1;
    aPL0 += aPL1;

    __shared__ float sm[4][3][256];        // 12 KB: per-wave partial 16x16 f32 accs
#pragma unroll
    for (int v = 0; v < 8; ++v) {
        sm[wave][0][lane * 8 + v] = aLL0[v];
        sm[wave][1][lane * 8 + v] = aPP0[v];
        sm[wave][2][lane * 8 + v] = aPL0[v];
    }
    __syncthreads();

    if (threadIdx.x == 0) {
        const float Minv = 1.f / (float)MTOT;
        // C/D layout: element (row d, col e) lives in lane e + 16*(d>=8), vgpr d%8
        auto S = [&](int mat, int d, int e) -> float {
            int p = (e + ((d >= 8) ? 16 : 0)) * 8 + (d & 7);
            float s = 0.f;
            for (int w = 0; w < 4; ++w) s += sm[w][mat][p];
            return s;
        };
        float meanL[9], meanP[9];
        for (int d = 0; d < 9; ++d) {
            meanL[d] = S(0, d, 9) * Minv;   // Σ la_d  (ones row)
            meanP[d] = S(1, d, 9) * Minv;   // Σ pr_d
        }
        float covLL[9][9], covPP[9][9], covPL[9][9];
        for (int d = 0; d < 9; ++d)
            for (int e = 0; e < 9; ++e) {
                covLL[d][e] = S(0, d, e) * Minv - meanL[d] * meanL[e];
                covPP[d][e] = S(1, d, e) * Minv - meanP[d] * meanP[e];
                covPL[d][e] = S(2, d, e) * Minv - meanP[d] * meanL[e];
            }
        // Ginv = inv(covLL + 1e-5 I) by Gauss-Jordan (SPD + ridge: no pivoting)
        float G[9][18];
        for (int a = 0; a < 9; ++a)
            for (int b = 0; b < 9; ++b) {
                G[a][b]     = covLL[a][b] + ((a == b) ? 1e-5f : 0.f);
                G[a][9 + b] = (a == b) ? 1.f : 0.f;
            }
        for (int k = 0; k < 9; ++k) {
            float ip = 1.f / G[k][k];
            for (int b = 0; b < 18; ++b) G[k][b] *= ip;
            for (int a = 0; a < 9; ++a) {
                if (a == k) continue;
                float f = G[a][k];
                for (int b = 0; b < 18; ++b) G[a][b] -= f * G[k][b];
            }
        }
        // V = covPP - covPL * Ginv * covPLᵀ + 1e-6 I
        float T[9][9];
        for (int d = 0; d < 9; ++d)
            for (int f = 0; f < 9; ++f) {
                float s = 0.f;
                for (int e = 0; e < 9; ++e) s += covPL[d][e] * G[e][9 + f];
                T[d][f] = s;
            }
        float V[9][9];
        for (int d = 0; d < 9; ++d)
            for (int g = 0; g < 9; ++g) {
                float s = 0.f;
                for (int f = 0; f < 9; ++f) s += T[d][f] * covPL[g][f];
                V[d][g] = covPP[d][g] - s + ((d == g) ? 1e-6f : 0.f);
            }
        // Cholesky log-det: 0.5 * 2 * Σ log(diag + 1e-8)
        float L[9][9];
        float ld = 0.f;
        for (int k = 0; k < 9; ++k) {
            float s = V[k][k];
            for (int b = 0; b < k; ++b) s -= L[k][b] * L[k][b];
            float dk = sqrtf(fmaxf(s, 0.f));
            ld += logf(dk + 1e-8f);
            float inv = 1.f / fmaxf(dk, 1e-20f);
            for (int a = k + 1; a < 9; ++a) {
                float t = V[a][k];
                for (int b = 0; b < k; ++b) t -= L[a][b] * L[k][b];
                L[a][k] = t * inv;
            }
            L[k][k] = dk;
        }
        results[nc] = ld;
    }
}

// ---------------------------------------------------------------------------
// K3: deterministic fixed-order reduction of the 76 per-(n,c) values.
// mean over N then sum/NUM_CLASSES  ==  Σ / (N*NUM_CLASSES); LOSS_WEIGHT = 1.
// ---------------------------------------------------------------------------
__global__ void rmi_reduce(const float* __restrict__ results,
                           float*       __restrict__ out)
{
    if (threadIdx.x == 0 && blockIdx.x == 0) {
        float s = 0.f;
        for (int a = 0; a < NCPROB; ++a) s += results[a];
        out[0] = s * (1.0f / (float)NCPROB);
    }
}

extern "C" void kernel_launch(void* const* d_in, const int* in_sizes, int n_in,
                              void* d_out, int out_size, void* d_ws, size_t ws_size,
                              hipStream_t stream)
{
    (void)in_sizes; (void)n_in; (void)out_size; (void)ws_size;
    const float* logits = (const float*)d_in[0];   // (4,19,512,512) f32
    const int*   label  = (const int*)d_in[1];     // (4,512,512) int

    float* pooledP = (float*)d_ws;                                   // 4*19*171*171 f32
    float* pooledO = pooledP + (size_t)NB * NCLS * PH * PH;          // same size
    float* results = pooledO + (size_t)NB * NCLS * PH * PH;          // 76 f32

    const int total = NB * PH * PH;
    rmi_softmax_pool<<<(total + 255) / 256, 256, 0, stream>>>(logits, label, pooledP, pooledO);
    rmi_cov_wmma<<<NCPROB, 128, 0, stream>>>(pooledP, pooledO, results);
    rmi_reduce<<<1, 32, 0, stream>>>(results, (float*)d_out);
}